// VolumeCarver_11940009083335
// MI455X (gfx1250) — compile-verified
//
#include <hip/hip_runtime.h>
#include <hip/hip_bf16.h>
#include <stdint.h>

// Volume rendering weight-sum, MI455X (gfx1250).
// One ray per thread; 128-sample serial scan per thread; rows staged
// global->LDS with async-to-LDS B128 copies (ASYNCcnt pipeline, depth 2),
// logits gathers issued one chunk ahead. Pure bandwidth-bound: ~260 MiB
// => ~11 us floor at 23.3 TB/s.

#define LEN      128
#define CHUNK    4
#define NCHUNK   (LEN / CHUNK)   // 32 chunks of float4/int4 per ray
#define BLOCK    256             // 8 wave32 per block
#define NBUF     4               // LDS ring buffers (3 live: read j, j+1; write j+2)
#define MAX_DIST 1e10f
#define EPS_W    1e-10f

__device__ __forceinline__ float fast_sigmoid(const float x) {
    return __builtin_amdgcn_rcpf(1.0f + __expf(-x));
}

__global__ __launch_bounds__(BLOCK) void VolumeCarver_kernel(
    const float* __restrict__ t_stops,   // [R, 128]
    const int*   __restrict__ leaves,    // [R, 128]
    const float* __restrict__ logits,    // [1M]
    float*       __restrict__ out,       // [R]
    int num_rays)
{
    __shared__ float4 lds_t[NBUF * BLOCK];   // 16 KB
    __shared__ int4   lds_l[NBUF * BLOCK];   // 16 KB

    const int tid = threadIdx.x;
    const int ray = blockIdx.x * BLOCK + tid;
    if (ray >= num_rays) return;

    const float* tbase = t_stops + (size_t)ray * LEN;
    const int*   lbase = leaves  + (size_t)ray * LEN;

    // LDS byte addresses: low 32 bits of the flat shared pointer are the LDS
    // offset (LDS aperture has zero low bits per the gfx1250 aperture map).
    unsigned ldsT[NBUF], ldsL[NBUF];
#pragma unroll
    for (int b = 0; b < NBUF; ++b) {
        ldsT[b] = (unsigned)(uintptr_t)&lds_t[b * BLOCK + tid];
        ldsL[b] = (unsigned)(uintptr_t)&lds_l[b * BLOCK + tid];
    }

    // Issue one chunk (16B t_stops + 16B leaves per lane) as two async
    // global->LDS B128 copies. ASYNCcnt += 2 per call; completes in order.
    auto issue_chunk = [&](int c) {
        const unsigned lt = ldsT[c & (NBUF - 1)];
        const unsigned ll = ldsL[c & (NBUF - 1)];
        const unsigned long long gt = (unsigned long long)(uintptr_t)(tbase + c * CHUNK);
        const unsigned long long gl = (unsigned long long)(uintptr_t)(lbase + c * CHUNK);
        asm volatile(
            "global_load_async_to_lds_b128 %0, %2, off\n\t"
            "global_load_async_to_lds_b128 %1, %3, off"
            :: "v"(lt), "v"(ll), "v"(gt), "v"(gl)
            : "memory");
    };

    // ---- prologue: fill pipeline with chunks 0 and 1 ----
    issue_chunk(0);
    issue_chunk(1);
    asm volatile("s_wait_asynccnt 2" ::: "memory");   // chunk 0 resident

    // Pre-gather opacities for chunk 0 (indices now in LDS).
    int4 lv = lds_l[0 * BLOCK + tid];
    float oc0 = fast_sigmoid(logits[lv.x]);
    float oc1 = fast_sigmoid(logits[lv.y]);
    float oc2 = fast_sigmoid(logits[lv.z]);
    float oc3 = fast_sigmoid(logits[lv.w]);

    float T   = 1.0f;   // running cumprod of (shifted) transmittance
    float acc = 0.0f;   // weight sum

    for (int j = 0; j < NCHUNK; ++j) {
        // Keep two chunks in flight; wait until chunk j+1 has landed.
        if (j + 2 < NCHUNK) {
            issue_chunk(j + 2);
            asm volatile("s_wait_asynccnt 2" ::: "memory");
        } else {
            asm volatile("s_wait_asynccnt 0" ::: "memory");
        }

        const int bj = j & (NBUF - 1);
        const float4 t4 = lds_t[bj * BLOCK + tid];

        float tn = 0.0f;
        float on0 = 0.0f, on1 = 0.0f, on2 = 0.0f, on3 = 0.0f;
        const bool has_next = (j + 1 < NCHUNK);
        if (has_next) {
            const int bn = (j + 1) & (NBUF - 1);
            tn = lds_t[bn * BLOCK + tid].x;
            // Prefetch next chunk's opacity gathers a full chunk early.
            const int4 lvn = lds_l[bn * BLOCK + tid];
            on0 = fast_sigmoid(logits[lvn.x]);
            on1 = fast_sigmoid(logits[lvn.y]);
            on2 = fast_sigmoid(logits[lvn.z]);
            on3 = fast_sigmoid(logits[lvn.w]);
        }

        const float d0 = t4.y - t4.x;
        const float d1 = t4.z - t4.y;
        const float d2 = t4.w - t4.z;
        const float d3 = has_next ? (tn - t4.w) : MAX_DIST;

        // element 4j+0 : global element 0 has trans_shifted == 1
        {
            const float a  = 1.0f - __expf(-oc0 * d0);
            const float tr = fminf(1.0f, 1.0f - a + EPS_W);
            T = (j == 0) ? T : T * tr;
            acc = fmaf(a, T, acc);
        }
        // elements 4j+1..3
        {
            const float a  = 1.0f - __expf(-oc1 * d1);
            const float tr = fminf(1.0f, 1.0f - a + EPS_W);
            T *= tr;  acc = fmaf(a, T, acc);
        }
        {
            const float a  = 1.0f - __expf(-oc2 * d2);
            const float tr = fminf(1.0f, 1.0f - a + EPS_W);
            T *= tr;  acc = fmaf(a, T, acc);
        }
        {
            const float a  = 1.0f - __expf(-oc3 * d3);
            const float tr = fminf(1.0f, 1.0f - a + EPS_W);
            T *= tr;  acc = fmaf(a, T, acc);
        }

        oc0 = on0; oc1 = on1; oc2 = on2; oc3 = on3;
    }

    out[ray] = acc;
}

extern "C" void kernel_launch(void* const* d_in, const int* in_sizes, int n_in,
                              void* d_out, int out_size, void* d_ws, size_t ws_size,
                              hipStream_t stream) {
    (void)n_in; (void)out_size; (void)d_ws; (void)ws_size;
    const float* t_stops = (const float*)d_in[0];
    const int*   leaves  = (const int*)d_in[1];
    const float* logits  = (const float*)d_in[2];
    float*       out     = (float*)d_out;

    const int num_rays = in_sizes[0] / LEN;        // 262144
    const int grid     = (num_rays + BLOCK - 1) / BLOCK;

    VolumeCarver_kernel<<<grid, BLOCK, 0, stream>>>(t_stops, leaves, logits, out, num_rays);
}